// PointnetFeatures_19318762897460
// MI455X (gfx1250) — compile-verified
//
#include <hip/hip_runtime.h>
#include <hip/hip_bf16.h>

typedef __attribute__((ext_vector_type(2))) float    v2f;
typedef __attribute__((ext_vector_type(8))) float    v8f;
typedef __attribute__((ext_vector_type(4))) unsigned v4u;
typedef __attribute__((ext_vector_type(8))) unsigned v8u;

#define LSEQ  1024
#define BATCH 32
#define MROWS (BATCH * LSEQ)   // 32768
#define SDIM  16
#define KCONV 4
#define EPSV  1e-5f

__device__ __forceinline__ float sigmoidf_(float x) { return 1.0f / (1.0f + __expf(-x)); }

// ---------------------------------------------------------------------------
// Tensor Data Mover: DMA a 2D f32 tile [tile_d1 rows x tile_d0 cols] from a
// row-major tensor (row stride = stride_d0 elements) into LDS at lds_byte_off,
// inserting 1 dword of padding after every 16 dwords (pad_interval code 3,
// pad_amount code 0) => LDS row stride of 17 floats (bank-conflict-free).
// D# per CDNA5 ISA: group0 = count|lds|global|type=2,
// group1 = data_size(4B)|pad|dims|tile|stride. 2D => VADDR2/3 omitted.
// ---------------------------------------------------------------------------
__device__ __forceinline__ void tdm_load_2d(unsigned lds_byte_off, const float* gptr,
                                            unsigned tensor_d0, unsigned tensor_d1,
                                            unsigned tile_d0, unsigned tile_d1,
                                            unsigned stride_d0)
{
    unsigned long long ga = (unsigned long long)(uintptr_t)gptr;
    v4u g0;
    g0[0] = 1u;                                       // count=1, user descriptor
    g0[1] = lds_byte_off;                             // lds_addr (bytes)
    g0[2] = (unsigned)(ga & 0xFFFFFFFFu);             // global_addr[31:0]
    g0[3] = (unsigned)((ga >> 32) & 0x01FFFFFFu)      // global_addr[56:32]
          | 0x80000000u;                              // type = 2 ("image")
    v8u g1;
    g1[0] = (2u << 16)                                // data_size = 4 bytes
          | (1u << 20)                                // pad_enable
          | (3u << 22)                                // pad_interval: every 16 dwords
          | (0u << 25);                               // pad_amount: 1 dword
    g1[1] = (tensor_d0 & 0xFFFFu) << 16;              // tensor_dim0[15:0]
    g1[2] = (tensor_d0 >> 16) | ((tensor_d1 & 0xFFFFu) << 16);
    g1[3] = (tensor_d1 >> 16) | (tile_d0 << 16);      // tile_dim0
    g1[4] = tile_d1;                                  // tile_dim1 (tile_dim2 = 0)
    g1[5] = stride_d0;                                // tensor_dim0_stride[31:0]
    g1[6] = 0u;                                       // stride[47:32], dim1_stride lo
    g1[7] = 0u;
    asm volatile("tensor_load_to_lds %0, %1" :: "s"(g0), "s"(g1) : "memory");
}

// ---------------------------------------------------------------------------
// WMMA fp32 GEMM:  Y[M,N] = X[M,K] @ W[N,K]^T + bias[N]
// BM=128, BN=64, BK=16. 256 threads = 8 waves; wave w owns rows [w*16, w*16+16),
// 4 accumulators cover 64 columns. V_WMMA_F32_16X16X4_F32 (exact fp32).
// Tiles staged by TDM with double-buffered ping-pong: DMA of tile k+1 overlaps
// WMMA on tile k; TENSORcnt + workgroup barrier order DMA writes vs DS reads.
// A fragment (16x4 f32, ISA 7.12.2): lane l(0-15) row M=l, v0=K(2*half), v1=+1.
// C/D fragment: VGPR r -> M = r + 8*half, N = lane&15.
// ---------------------------------------------------------------------------
__global__ __launch_bounds__(256) void wmma_gemm_bias(
    const float* __restrict__ X, const float* __restrict__ W,
    const float* __restrict__ bias, float* __restrict__ Y, int N, int K)
{
    __shared__ float As[2][128][17];
    __shared__ float Bs[2][64][17];

    const int tid  = threadIdx.x;
    const int wave = tid >> 5;
    const int lane = tid & 31;
    const int half = lane >> 4;
    const int l16  = lane & 15;
    const int m0   = blockIdx.y * 128;
    const int n0   = blockIdx.x * 64;

    const unsigned ldsA0 = (unsigned)(uintptr_t)(&As[0][0][0]);
    const unsigned ldsA1 = (unsigned)(uintptr_t)(&As[1][0][0]);
    const unsigned ldsB0 = (unsigned)(uintptr_t)(&Bs[0][0][0]);
    const unsigned ldsB1 = (unsigned)(uintptr_t)(&Bs[1][0][0]);

    v8f acc[4] = {};

    if (tid == 0) {     // prologue DMA: first K-tile into buffer 0
        tdm_load_2d(ldsA0, X + (size_t)m0 * K, (unsigned)K, (unsigned)(MROWS - m0),
                    16u, 128u, (unsigned)K);
        tdm_load_2d(ldsB0, W + (size_t)n0 * K, (unsigned)K, (unsigned)(N - n0),
                    16u, 64u, (unsigned)K);
    }

    int buf = 0;
    for (int k0 = 0; k0 < K; k0 += 16) {
        __builtin_amdgcn_s_wait_tensorcnt(0);   // every wave drains its own TDM ops
        __syncthreads();                        // tile `buf` visible to all waves
        if (tid == 0 && (k0 + 16) < K) {        // prefetch next tile into other buffer
            const unsigned la = buf ? ldsA0 : ldsA1;
            const unsigned lb = buf ? ldsB0 : ldsB1;
            tdm_load_2d(la, X + (size_t)m0 * K + (k0 + 16), (unsigned)(K - k0 - 16),
                        (unsigned)(MROWS - m0), 16u, 128u, (unsigned)K);
            tdm_load_2d(lb, W + (size_t)n0 * K + (k0 + 16), (unsigned)(K - k0 - 16),
                        (unsigned)(N - n0), 16u, 64u, (unsigned)K);
        }
        #pragma unroll
        for (int k4 = 0; k4 < 16; k4 += 4) {
            v2f a;
            a.x = As[buf][wave * 16 + l16][k4 + 2 * half];
            a.y = As[buf][wave * 16 + l16][k4 + 2 * half + 1];
            #pragma unroll
            for (int nt = 0; nt < 4; ++nt) {
                v2f b;
                b.x = Bs[buf][nt * 16 + l16][k4 + 2 * half];
                b.y = Bs[buf][nt * 16 + l16][k4 + 2 * half + 1];
                acc[nt] = __builtin_amdgcn_wmma_f32_16x16x4_f32(
                    false, a, false, b, (short)0, acc[nt], false, false);
            }
        }
        __syncthreads();    // emits s_wait_dscnt 0 + barrier: all reads of `buf`
        buf ^= 1;           // complete before its next DMA overwrite
    }

    #pragma unroll
    for (int nt = 0; nt < 4; ++nt) {
        const int col = n0 + nt * 16 + l16;
        const float bv = bias[col];
        #pragma unroll
        for (int r = 0; r < 8; ++r) {
            const int row = m0 + wave * 16 + half * 8 + r;
            Y[(size_t)row * N + col] = acc[nt][r] + bv;
        }
    }
}

// Small-K GEMM (layer-1 linear, K=3): one thread per (m,n).
__global__ __launch_bounds__(256) void small_gemm_bias(
    const float* __restrict__ X, const float* __restrict__ W,
    const float* __restrict__ bias, float* __restrict__ Y, int N, int K)
{
    size_t idx = (size_t)blockIdx.x * 256 + threadIdx.x;   // over M*N
    int n = (int)(idx % N); size_t m = idx / N;
    float acc = bias[n];
    for (int k = 0; k < K; ++k) acc += X[m * K + k] * W[(size_t)n * K + k];
    Y[idx] = acc;
}

// [B,L,C] row-major -> [B,C,L]
__global__ __launch_bounds__(256) void transpose_blc(
    const float* __restrict__ in, float* __restrict__ out, int C)
{
    size_t idx = (size_t)blockIdx.x * 256 + threadIdx.x;   // over M*C
    int c = (int)(idx % C); size_t m = idx / C;
    int b = (int)(m / LSEQ), l = (int)(m % LSEQ);
    out[((size_t)b * C + c) * LSEQ + l] = in[idx];
}

// Per-channel batch mean/var (biased) over all M rows. One block per channel.
__global__ __launch_bounds__(256) void bn_stats(
    const float* __restrict__ a, float* __restrict__ mu, float* __restrict__ var, int C)
{
    __shared__ float r1[256], r2[256];
    const int c = blockIdx.x, tid = threadIdx.x;
    float s = 0.f, sq = 0.f;
    for (int m = tid; m < MROWS; m += 256) {
        float v = a[(size_t)m * C + c];
        s += v; sq += v * v;
    }
    r1[tid] = s; r2[tid] = sq; __syncthreads();
    for (int o = 128; o > 0; o >>= 1) {
        if (tid < o) { r1[tid] += r1[tid + o]; r2[tid] += r2[tid + o]; }
        __syncthreads();
    }
    if (tid == 0) {
        float me = r1[0] / (float)MROWS;
        mu[c] = me;
        var[c] = r2[0] / (float)MROWS - me * me;
    }
}

// bn = (a-mu)*rsqrt(var+eps)*g + be ; write bn transposed to d_out, relu to r (row-major)
__global__ __launch_bounds__(256) void bn_relu(
    const float* __restrict__ a, const float* __restrict__ mu, const float* __restrict__ var,
    const float* __restrict__ g, const float* __restrict__ be,
    float* __restrict__ r, float* __restrict__ bn_t, int C)
{
    size_t idx = (size_t)blockIdx.x * 256 + threadIdx.x;   // over M*C
    int c = (int)(idx % C); size_t m = idx / C;
    float v = (a[idx] - mu[c]) * rsqrtf(var[c] + EPSV) * g[c] + be[c];
    int b = (int)(m / LSEQ), l = (int)(m % LSEQ);
    bn_t[((size_t)b * C + c) * LSEQ + l] = v;
    r[idx] = fmaxf(v, 0.f);
}

// causal depthwise conv (K=4, left-pad 3, correlation orientation) + SiLU
__global__ __launch_bounds__(256) void conv_silu(
    const float* __restrict__ r, const float* __restrict__ w,
    float* __restrict__ xc, int C)
{
    size_t idx = (size_t)blockIdx.x * 256 + threadIdx.x;   // over M*C
    int c = (int)(idx % C); size_t m = idx / C;
    int l = (int)(m % LSEQ); size_t brow = m - l;
    float acc = 0.f;
    #pragma unroll
    for (int j = 0; j < KCONV; ++j) {
        int ls = l - (KCONV - 1) + j;
        if (ls >= 0) acc += w[c * KCONV + j] * r[(brow + ls) * C + c];
    }
    xc[idx] = acc * sigmoidf_(acc);
}

// Bm, Cm, delta, Bbar projections (S=16). One thread per (m,s).
__global__ __launch_bounds__(256) void proj_bcd(
    const float* __restrict__ xp, const float* __restrict__ Bw, const float* __restrict__ Cw,
    float* __restrict__ dlt, float* __restrict__ Bb, float* __restrict__ Cc, int C)
{
    size_t idx = (size_t)blockIdx.x * 256 + threadIdx.x;   // over M*16
    int s = (int)(idx & 15); size_t m = idx >> 4;
    const float* xr = xp + m * C;
    float ab = 0.f, ac = 0.f;
    for (int k = 0; k < C; ++k) {
        float x = xr[k];
        ab += x * Bw[(size_t)s * C + k];
        ac += x * Cw[(size_t)s * C + k];
    }
    float d = sigmoidf_(xr[s]);
    dlt[idx] = d;
    Bb[idx]  = ab * d;
    Cc[idx]  = ac;
}

__global__ void neg_exp_k(const float* __restrict__ A, float* __restrict__ An, int n)
{
    int i = blockIdx.x * 64 + threadIdx.x;
    if (i < n) An[i] = -__expf(A[i]);
}

// Selective scan: one thread per (b,c); 16-state recurrence in registers.
// Per-timestep (delta,Bbar,C) shared across channels -> staged in LDS,
// broadcast-read (conflict-free). xp D-term load is coalesced across lanes.
__global__ __launch_bounds__(64) void scan_kernel(
    const float* __restrict__ An,  const float* __restrict__ dlt,
    const float* __restrict__ Bb,  const float* __restrict__ Cc,
    const float* __restrict__ xp,  const float* __restrict__ Dv,
    float* __restrict__ y, int C)
{
    __shared__ float sD[32 * 16], sB[32 * 16], sC[32 * 16];
    const int b = blockIdx.x;
    const int c = blockIdx.y * 64 + threadIdx.x;
    float h[16], Ar[16];
    #pragma unroll
    for (int s = 0; s < 16; ++s) { h[s] = 0.f; Ar[s] = An[(size_t)c * 16 + s]; }
    const float Dc = Dv[c];

    for (int l0 = 0; l0 < LSEQ; l0 += 32) {
        __syncthreads();
        const size_t base = ((size_t)b * LSEQ + l0) * 16;
        for (int i = threadIdx.x; i < 512; i += 64) {
            sD[i] = dlt[base + i]; sB[i] = Bb[base + i]; sC[i] = Cc[base + i];
        }
        __syncthreads();
        for (int t = 0; t < 32; ++t) {
            float yv = 0.f;
            #pragma unroll
            for (int s = 0; s < 16; ++s) {
                float e = __expf(Ar[s] * sD[t * 16 + s]);
                h[s] = e * h[s] + sB[t * 16 + s];
                yv += h[s] * sC[t * 16 + s];
            }
            const size_t m = (size_t)b * LSEQ + l0 + t;
            y[m * C + c] = yv + Dc * xp[m * C + c];
        }
    }
}

// LayerNorm(out + residual) * g + be ; write x_next row-major, mask transposed.
__global__ __launch_bounds__(256) void ln_residual_mask(
    const float* __restrict__ out, const float* __restrict__ res,
    const float* __restrict__ g,  const float* __restrict__ be,
    float* __restrict__ xn, float* __restrict__ mask_t, int C)
{
    __shared__ float red[256];
    const int m = blockIdx.x, tid = threadIdx.x;
    float s = 0.f, sq = 0.f;
    for (int c = tid; c < C; c += 256) {
        float v = out[(size_t)m * C + c] + res[(size_t)m * C + c];
        s += v; sq += v * v;
    }
    red[tid] = s; __syncthreads();
    for (int o = 128; o > 0; o >>= 1) { if (tid < o) red[tid] += red[tid + o]; __syncthreads(); }
    const float mean = red[0] / (float)C;
    __syncthreads();
    red[tid] = sq; __syncthreads();
    for (int o = 128; o > 0; o >>= 1) { if (tid < o) red[tid] += red[tid + o]; __syncthreads(); }
    const float var = red[0] / (float)C - mean * mean;
    const float inv = rsqrtf(var + EPSV);
    const int b = m / LSEQ, l = m % LSEQ;
    for (int c = tid; c < C; c += 256) {
        float v = out[(size_t)m * C + c] + res[(size_t)m * C + c];
        float x = (v - mean) * inv * g[c] + be[c];
        xn[(size_t)m * C + c] = x;
        mask_t[((size_t)b * C + c) * LSEQ + l] = (x > 0.f) ? 1.f : 0.f;
    }
}

// Global max-pool over points: feat[b,c] = max_n x3[b,n,c], C=1024.
__global__ __launch_bounds__(256) void feat_max(
    const float* __restrict__ x, float* __restrict__ feat)
{
    int idx = blockIdx.x * 256 + threadIdx.x;   // 32*1024
    int c = idx & 1023, b = idx >> 10;
    float mx = -INFINITY;
    for (int n = 0; n < LSEQ; ++n)
        mx = fmaxf(mx, x[((size_t)b * LSEQ + n) * 1024 + c]);
    feat[idx] = mx;
}

// ---------------------------------------------------------------------------
struct LayerP {
    const float *w, *b, *g, *be, *conv_w, *A, *D, *Bw, *Cw,
                *in_w, *in_b, *out_w, *out_b, *ln_g, *ln_b;
};

static LayerP get_layer(void* const* d_in, int base)
{
    LayerP p;
    p.w      = (const float*)d_in[base + 0];
    p.b      = (const float*)d_in[base + 1];
    p.g      = (const float*)d_in[base + 2];
    p.be     = (const float*)d_in[base + 3];
    p.conv_w = (const float*)d_in[base + 4];
    p.A      = (const float*)d_in[base + 5];
    p.D      = (const float*)d_in[base + 6];
    p.Bw     = (const float*)d_in[base + 7];
    p.Cw     = (const float*)d_in[base + 8];
    p.in_w   = (const float*)d_in[base + 9];
    p.in_b   = (const float*)d_in[base + 10];
    p.out_w  = (const float*)d_in[base + 11];
    p.out_b  = (const float*)d_in[base + 12];
    p.ln_g   = (const float*)d_in[base + 13];
    p.ln_b   = (const float*)d_in[base + 14];
    return p;
}

// s1/s2/s3 are M*1024 scratch buffers; s3 may alias x_in (x_in dead after GEMM1).
// Result x_next lands in s1.
static void run_layer(hipStream_t st, const float* x_in, int Cin, int C, const LayerP& p,
                      float* s1, float* s2, float* s3,
                      float* dlt, float* Bb, float* Cc, float* An, float* mu, float* var,
                      float* a_t, float* bn_t, float* mask_t)
{
    const int elems = MROWS * C;
    if (Cin >= 16)
        wmma_gemm_bias<<<dim3(C / 64, MROWS / 128), 256, 0, st>>>(x_in, p.w, p.b, s1, C, Cin);
    else
        small_gemm_bias<<<elems / 256, 256, 0, st>>>(x_in, p.w, p.b, s1, C, Cin);
    transpose_blc<<<elems / 256, 256, 0, st>>>(s1, a_t, C);
    bn_stats<<<C, 256, 0, st>>>(s1, mu, var, C);
    bn_relu<<<elems / 256, 256, 0, st>>>(s1, mu, var, p.g, p.be, s2, bn_t, C);
    conv_silu<<<elems / 256, 256, 0, st>>>(s2, p.conv_w, s1, C);
    wmma_gemm_bias<<<dim3(C / 64, MROWS / 128), 256, 0, st>>>(s1, p.in_w, p.in_b, s3, C, C);
    proj_bcd<<<(MROWS * SDIM) / 256, 256, 0, st>>>(s3, p.Bw, p.Cw, dlt, Bb, Cc, C);
    neg_exp_k<<<(C * SDIM + 63) / 64, 64, 0, st>>>(p.A, An, C * SDIM);
    scan_kernel<<<dim3(BATCH, C / 64), 64, 0, st>>>(An, dlt, Bb, Cc, s3, p.D, s1, C);
    wmma_gemm_bias<<<dim3(C / 64, MROWS / 128), 256, 0, st>>>(s1, p.out_w, p.out_b, s3, C, C);
    ln_residual_mask<<<MROWS, 256, 0, st>>>(s3, s2, p.ln_g, p.ln_b, s1, mask_t, C);
}

extern "C" void kernel_launch(void* const* d_in, const int* in_sizes, int n_in,
                              void* d_out, int out_size, void* d_ws, size_t ws_size,
                              hipStream_t stream)
{
    (void)in_sizes; (void)n_in; (void)out_size; (void)ws_size;
    const float* points = (const float*)d_in[0];
    LayerP p1 = get_layer(d_in, 1);
    LayerP p2 = get_layer(d_in, 16);
    LayerP p3 = get_layer(d_in, 31);

    float* out = (float*)d_out;
    // output offsets (floats): feat, M1, M2, M3, a1, a2, a3, bn1, bn2, bn3
    const size_t SZ1 = (size_t)BATCH * 64 * LSEQ;     // 2,097,152
    const size_t SZ2 = (size_t)BATCH * 128 * LSEQ;    // 4,194,304
    const size_t SZ3 = (size_t)BATCH * 1024 * LSEQ;   // 33,554,432
    const size_t o_feat = 0;
    const size_t o_m1 = o_feat + (size_t)BATCH * 1024;
    const size_t o_m2 = o_m1 + SZ1;
    const size_t o_m3 = o_m2 + SZ2;
    const size_t o_a1 = o_m3 + SZ3;
    const size_t o_a2 = o_a1 + SZ1;
    const size_t o_a3 = o_a2 + SZ2;
    const size_t o_b1 = o_a3 + SZ3;
    const size_t o_b2 = o_b1 + SZ1;
    const size_t o_b3 = o_b2 + SZ2;

    float* ws = (float*)d_ws;
    const size_t big = (size_t)MROWS * 1024;
    float* P0  = ws;
    float* P1  = ws + big;
    float* P2  = ws + 2 * big;
    float* dlt = ws + 3 * big;
    float* Bb  = dlt + (size_t)MROWS * SDIM;
    float* Cc  = Bb  + (size_t)MROWS * SDIM;
    float* An  = Cc  + (size_t)MROWS * SDIM;
    float* mu  = An  + 1024 * SDIM;
    float* var = mu  + 1024;

    // Layer 1: x_in = points, scratch rotation (P0,P1,P2) -> x1 in P0
    run_layer(stream, points, 3, 64, p1, P0, P1, P2,
              dlt, Bb, Cc, An, mu, var, out + o_a1, out + o_b1, out + o_m1);
    // Layer 2: x_in = P0, scratch (P1,P2,P0) -> x2 in P1
    run_layer(stream, P0, 64, 128, p2, P1, P2, P0,
              dlt, Bb, Cc, An, mu, var, out + o_a2, out + o_b2, out + o_m2);
    // Layer 3: x_in = P1, scratch (P0,P2,P1) -> x3 in P0
    run_layer(stream, P1, 128, 1024, p3, P0, P2, P1,
              dlt, Bb, Cc, An, mu, var, out + o_a3, out + o_b3, out + o_m3);

    feat_max<<<(BATCH * 1024) / 256, 256, 0, stream>>>(P0, out + o_feat);
}